// MGDCF_66967130079601
// MI455X (gfx1250) — compile-verified
//
#include <hip/hip_runtime.h>
#include <cstdint>

// ---------------------------------------------------------------------------
// MGDCF propagation: h_{k+1} = beta * (A_norm h_k) + alpha * h_0, K=4, D=64.
// Memory-bound sparse scatter/gather; working set fits MI455X 192MB L2.
// CDNA5 paths: global_load_async_to_lds_b128 (+ s_wait_asynccnt) staging,
// global_atomic_add_f32 scatter, global_load_b128 gathers, global_prefetch_b8.
// ---------------------------------------------------------------------------

#ifndef __has_builtin
#define __has_builtin(x) 0
#endif

#if __has_builtin(__builtin_amdgcn_global_load_async_to_lds_b128)
#define HAVE_ASYNC_LDS 1
#else
#define HAVE_ASYNC_LDS 0
#endif

#if HAVE_ASYNC_LDS
// Builtin signature (from hipcc diagnostic): arg0 = int4* in AS1 (global),
// arg1 = int4* in AS3 (LDS), arg2/arg3 = imm offset / imm cpol.
typedef __attribute__((__vector_size__(4 * sizeof(int)))) int i32x4;
typedef __attribute__((address_space(1))) i32x4* gv4_ptr;
typedef __attribute__((address_space(3))) i32x4* lv4_ptr;
#if __has_builtin(__builtin_amdgcn_s_wait_asynccnt)
#define WAIT_ASYNC(n) __builtin_amdgcn_s_wait_asynccnt(n)
#else
#define WAIT_ASYNC(n) asm volatile("s_wait_asynccnt %0" ::"i"(n) : "memory")
#endif
#endif

#define WG 256
static constexpr int D               = 64;  // EMBED_DIM (reference constant)
static constexpr int LANES_PER_EDGE  = 16;  // 16 lanes x float4 = 64 floats
static constexpr int BATCH           = WG / LANES_PER_EDGE;  // 16 edges/batch
static constexpr int NBATCH          = 8;                    // batches/block
static constexpr int EDGES_PER_BLOCK = BATCH * NBATCH;       // 128 edges

__device__ __forceinline__ void atomic_add_f32(float* p, float v) {
  (void)__hip_atomic_fetch_add(p, v, __ATOMIC_RELAXED, __HIP_MEMORY_SCOPE_AGENT);
}

// ---- degree ----------------------------------------------------------------
__global__ __launch_bounds__(WG) void k_zero(float* __restrict__ p, int n) {
  int i = blockIdx.x * blockDim.x + threadIdx.x;
  if (i < n) p[i] = 0.0f;
}

__global__ __launch_bounds__(WG) void k_degree(const int* __restrict__ dst,
                                               float* __restrict__ deg, int E) {
  int e = blockIdx.x * blockDim.x + threadIdx.x;
  if (e < E) atomic_add_f32(&deg[dst[e]], 1.0f);
}

// ---- per-edge weight: beta * rsqrt(deg[src]) * rsqrt(deg[dst]) -------------
__global__ __launch_bounds__(WG) void k_norm(const int* __restrict__ src,
                                             const int* __restrict__ dst,
                                             const float* __restrict__ deg,
                                             float* __restrict__ wnorm,
                                             float beta, int E) {
  int e = blockIdx.x * blockDim.x + threadIdx.x;
  if (e < E) {
    float ds = deg[src[e]];
    float dd = deg[dst[e]];
    float a = ds > 0.0f ? __frsqrt_rn(ds) : 0.0f;
    float b = dd > 0.0f ? __frsqrt_rn(dd) : 0.0f;
    wnorm[e] = beta * a * b;
  }
}

// ---- agg = coeff * h0  (folds alpha, and alpha/gamma on the last iter) -----
__global__ __launch_bounds__(WG) void k_init(const float4* __restrict__ x,
                                             float4* __restrict__ agg,
                                             float coeff, int n4) {
  int i = blockIdx.x * blockDim.x + threadIdx.x;
  if (i < n4) {
    float4 v = x[i];
    agg[i] = make_float4(coeff * v.x, coeff * v.y, coeff * v.z, coeff * v.w);
  }
}

// ---- scatter: agg[dst] += scale * wnorm[e] * h[src] ------------------------
__global__ __launch_bounds__(WG)
void k_scatter(const float* __restrict__ h, const int* __restrict__ src,
               const int* __restrict__ dst, const float* __restrict__ wnorm,
               float* __restrict__ agg, float scale, int E) {
  const int tid  = threadIdx.x;
  const int slot = tid >> 4;   // edge slot within batch   (0..15)
  const int lane = tid & 15;   // float4 chunk within row  (0..15)
  const int e0   = blockIdx.x * EDGES_PER_BLOCK;

#if HAVE_ASYNC_LDS
  // Double-buffered async gather of h[src] rows into LDS via the CDNA5
  // async-copy engine; atomics on batch b overlap the in-flight loads of b+1.
  __shared__ float buf[2][BATCH * D];

  auto issue = [&](int b, int pb) {
    int e = e0 + b * BATCH + slot;
    int s = (e < E) ? src[e] : 0;                       // clamp: safe dummy
    const float* gp = h + (size_t)s * D + lane * 4;
    float*       lp = &buf[pb][slot * D + lane * 4];
    __builtin_amdgcn_global_load_async_to_lds_b128(
        (gv4_ptr)(uintptr_t)gp, (lv4_ptr)(uint32_t)(uintptr_t)lp, 0, 0);
  };

  issue(0, 0);
  for (int b = 0; b < NBATCH; ++b) {
    const int pb = b & 1;
    if (b + 1 < NBATCH) {
      issue(b + 1, pb ^ 1);
      WAIT_ASYNC(1);          // batch b complete (async loads retire in order)
    } else {
      WAIT_ASYNC(0);
    }
    int e = e0 + b * BATCH + slot;
    if (e < E) {
      float w = scale * wnorm[e];
      int   d = dst[e];
      float4 v = *(const float4*)&buf[pb][slot * D + lane * 4];
      float* o = agg + (size_t)d * D + lane * 4;
      atomic_add_f32(o + 0, w * v.x);
      atomic_add_f32(o + 1, w * v.y);
      atomic_add_f32(o + 2, w * v.z);
      atomic_add_f32(o + 3, w * v.w);
    }
  }
#else
  for (int b = 0; b < NBATCH; ++b) {
    int e = e0 + b * BATCH + slot;
    if (e < E) {
      int en = e + BATCH;     // prefetch next batch's gather row
      if (en < E) __builtin_prefetch(h + (size_t)src[en] * D + lane * 4, 0, 0);
      int   s = src[e];
      int   d = dst[e];
      float w = scale * wnorm[e];
      float4 v = *(const float4*)(h + (size_t)s * D + lane * 4);
      float* o = agg + (size_t)d * D + lane * 4;
      atomic_add_f32(o + 0, w * v.x);
      atomic_add_f32(o + 1, w * v.y);
      atomic_add_f32(o + 2, w * v.z);
      atomic_add_f32(o + 3, w * v.w);
    }
  }
#endif
}

// ---------------------------------------------------------------------------
extern "C" void kernel_launch(void* const* d_in, const int* in_sizes, int n_in,
                              void* d_out, int out_size, void* d_ws, size_t ws_size,
                              hipStream_t stream) {
  const float* x   = (const float*)d_in[0];   // [N, 64] f32
  const int*   src = (const int*)d_in[1];     // [E] i32
  const int*   dst = (const int*)d_in[2];     // [E] i32

  const int ND = in_sizes[0];
  const int N  = ND / D;
  const int E  = in_sizes[1];
  float* out = (float*)d_out;

  // workspace layout: deg[N] | wnorm[E] | B0[N*D] | B1[N*D]   (~155 MB)
  float* ws    = (float*)d_ws;
  float* deg   = ws;
  float* wnorm = deg + N;
  float* B0    = wnorm + E;
  float* B1    = B0 + (size_t)ND;

  const double ALPHA = 0.1, BETA = 0.9;
  const int K = 4;
  double bk = 1.0, s = 0.0;
  for (int i = 0; i < K; ++i) { s += bk; bk *= BETA; }
  const double gamma = bk + ALPHA * s;        // == 1.0 for these constants
  const float  invg  = (float)(1.0 / gamma);

  const int n4 = ND / 4;
  const int gN  = (N  + WG - 1) / WG;
  const int gE  = (E  + WG - 1) / WG;
  const int g4  = (n4 + WG - 1) / WG;
  const int gSc = (E + EDGES_PER_BLOCK - 1) / EDGES_PER_BLOCK;

  k_zero  <<<gN, WG, 0, stream>>>(deg, N);
  k_degree<<<gE, WG, 0, stream>>>(dst, deg, E);
  k_norm  <<<gE, WG, 0, stream>>>(src, dst, deg, wnorm, (float)BETA, E);

  const float* hprev = x;
  float* pp[2] = {B0, B1};
  for (int k = 0; k < K; ++k) {
    const bool last = (k == K - 1);
    float* agg   = last ? out : pp[k & 1];
    float  coeff = last ? (float)(ALPHA / gamma) : (float)ALPHA;
    float  scale = last ? invg : 1.0f;
    k_init   <<<g4,  WG, 0, stream>>>((const float4*)x, (float4*)agg, coeff, n4);
    k_scatter<<<gSc, WG, 0, stream>>>(hprev, src, dst, wnorm, agg, scale, E);
    hprev = agg;
  }
}